// GCNNetwork_44375602102377
// MI455X (gfx1250) — compile-verified
//
#include <hip/hip_runtime.h>

#define LN_EPS 1e-5f

typedef __attribute__((ext_vector_type(2)))  float    v2f;
typedef __attribute__((ext_vector_type(8)))  float    v8f;

// ---------------------------------------------------------------- utilities
__global__ void fill_zero(float* __restrict__ p, long long n) {
  long long t = (long long)blockIdx.x * blockDim.x + threadIdx.x;
  if (t < n) p[t] = 0.0f;
}

__global__ void deg_count(const int* __restrict__ dstI, float* __restrict__ deg, int E) {
  int e = blockIdx.x * blockDim.x + threadIdx.x;
  if (e < E) atomicAdd(&deg[dstI[e]], 1.0f);
}

__global__ void deg_to_dinv(float* __restrict__ deg, int n) {
  int i = blockIdx.x * blockDim.x + threadIdx.x;
  if (i < n) deg[i] = rsqrtf(deg[i] + 1.0f);   // +1 self loop => deg >= 1
}

// ---------------------------------------------------------------- GEMM (WMMA)
// C[M x NO] = A[M x K] * W[K x NO], row-major fp32.
// One wave32 computes a 16-row x NO-col strip: A fragment loaded once per
// k-step and reused across NO/16 column tiles (8x less A traffic than
// tile-per-wave). W staged in LDS once per block; B fragments via ds_load.
template<int K, int NO, int WAVES>
__global__ void gemm_wmma(const float* __restrict__ A, const float* __restrict__ W,
                          float* __restrict__ C, int M) {
  __shared__ float Ws[K * NO];
  for (int i = threadIdx.x; i < K * NO; i += blockDim.x) Ws[i] = W[i];
  __syncthreads();

  const int lane = threadIdx.x & 31;
  const int wave = threadIdx.x >> 5;
  const int row0 = (blockIdx.x * WAVES + wave) * 16;
  if (row0 >= M) return;                       // wave-uniform: EXEC stays all-ones
  const int r  = lane & 15;
  const int hi = lane >> 4;

  constexpr int NT = NO / 16;
  v8f acc[NT] = {};

  // A 16x4 f32 fragment: lanes 0-15 -> K={k0,k0+1}, lanes 16-31 -> K={k0+2,k0+3}.
  const float* arow = A  + (size_t)(row0 + r) * K + 2 * hi;
  const float* wb   = Ws + r;

  #pragma unroll
  for (int k0 = 0; k0 < K; k0 += 4) {
    v2f a;
    a.x = arow[k0];
    a.y = arow[k0 + 1];
    #pragma unroll
    for (int j = 0; j < NT; ++j) {
      v2f b;                                    // B 4x16 f32 fragment (from LDS)
      b.x = wb[(k0 + 2 * hi)     * NO + 16 * j];
      b.y = wb[(k0 + 2 * hi + 1) * NO + 16 * j];
      acc[j] = __builtin_amdgcn_wmma_f32_16x16x4_f32(
          false, a, false, b, (short)0, acc[j], false, false);
    }
  }

  // D layout: VGPR i -> row = i + 8*hi, col = lane%16
  const int rbase = row0 + hi * 8;
  #pragma unroll
  for (int j = 0; j < NT; ++j) {
    #pragma unroll
    for (int i = 0; i < 8; ++i)
      C[(size_t)(rbase + i) * NO + 16 * j + r] = acc[j][i];
  }
}

// ---------------------------------------------------------------- aggregation
// One wave32 per edge (edges [0,E) real, [E,E+n) self-loops).
// Lane handles channels {lane, lane+32, ...} -> contiguous 128B per wave iter.
template<int F>
__global__ void aggregate(const float* __restrict__ H, float* __restrict__ OUT,
                          const int* __restrict__ srcI, const int* __restrict__ dstI,
                          const float* __restrict__ dinv, int E, int n) {
  long long t = (long long)blockIdx.x * blockDim.x + threadIdx.x;
  int e = (int)(t >> 5);
  if (e >= E + n) return;
  int lane = (int)(t & 31);
  int s, d;
  if (e < E) { s = srcI[e]; d = dstI[e]; } else { s = e - E; d = s; }
  float coef = dinv[s] * dinv[d];
  const float* hp = H   + (size_t)s * F;
  float*       op = OUT + (size_t)d * F;
  #pragma unroll
  for (int i = 0; i < F / 32; ++i) {
    int c = lane + 32 * i;
    atomicAdd(op + c, hp[c] * coef);           // global_atomic_add_f32 (L2-resident)
  }
}

// ---------------------------------------------------------------- bias + LN + ReLU
// One wave32 per node, 4 channels per lane, shfl-xor butterfly reductions.
__global__ void ln_relu(const float* __restrict__ AGG, const float* __restrict__ bias,
                        const float* __restrict__ g, const float* __restrict__ be,
                        float* __restrict__ OUT, int n) {
  long long t = (long long)blockIdx.x * blockDim.x + threadIdx.x;
  int node = (int)(t >> 5);
  if (node >= n) return;
  int lane = (int)(t & 31);
  float v[4];
  float sum = 0.0f;
  #pragma unroll
  for (int i = 0; i < 4; ++i) {
    int c = lane + 32 * i;
    v[i] = AGG[(size_t)node * 128 + c] + bias[c];
    sum += v[i];
  }
  #pragma unroll
  for (int off = 16; off >= 1; off >>= 1) sum += __shfl_xor(sum, off, 32);
  float mu = sum * (1.0f / 128.0f);
  float var = 0.0f;
  #pragma unroll
  for (int i = 0; i < 4; ++i) { float dd = v[i] - mu; var += dd * dd; }
  #pragma unroll
  for (int off = 16; off >= 1; off >>= 1) var += __shfl_xor(var, off, 32);
  var *= (1.0f / 128.0f);
  float rs = rsqrtf(var + LN_EPS);
  #pragma unroll
  for (int i = 0; i < 4; ++i) {
    int c = lane + 32 * i;
    float y = (v[i] - mu) * rs * g[c] + be[c];
    OUT[(size_t)node * 128 + c] = fmaxf(y, 0.0f);
  }
}

__global__ void add_bias64(float* __restrict__ out, const float* __restrict__ b, long long total) {
  long long t = (long long)blockIdx.x * blockDim.x + threadIdx.x;
  if (t < total) out[t] += b[t & 63];
}

// ---------------------------------------------------------------- launch
extern "C" void kernel_launch(void* const* d_in, const int* in_sizes, int n_in,
                              void* d_out, int out_size, void* d_ws, size_t ws_size,
                              hipStream_t stream) {
  const float* x   = (const float*)d_in[0];
  const int*   ei  = (const int*)  d_in[1];
  const float* W1  = (const float*)d_in[2];
  const float* b1  = (const float*)d_in[3];
  const float* g1v = (const float*)d_in[4];
  const float* be1 = (const float*)d_in[5];
  const float* W2  = (const float*)d_in[6];
  const float* b2  = (const float*)d_in[7];
  const float* g2v = (const float*)d_in[8];
  const float* be2 = (const float*)d_in[9];
  const float* W3  = (const float*)d_in[10];
  const float* b3  = (const float*)d_in[11];

  const int Nn = in_sizes[0] / 128;    // 100000
  const int E  = in_sizes[1] / 2;      // 1600000
  float* out = (float*)d_out;

  float* dinv = (float*)d_ws;          // N floats
  float* B0   = dinv + Nn;             // N x 128
  float* B1   = B0 + (size_t)Nn * 128; // N x 128

  const int* srcI = ei;
  const int* dstI = ei + E;

  const int BT = 256;
  auto cdiv = [](long long a, long long b) { return (unsigned)((a + b - 1) / b); };

  // --- symmetric normalization coefficients
  fill_zero  <<<cdiv(Nn, BT), BT, 0, stream>>>(dinv, Nn);
  deg_count  <<<cdiv(E,  BT), BT, 0, stream>>>(dstI, dinv, E);
  deg_to_dinv<<<cdiv(Nn, BT), BT, 0, stream>>>(dinv, Nn);

  dim3 blk(128);                       // 4 waves, 64 rows per block
  unsigned gemmBlocks = cdiv(Nn, 64);
  long long aggT = (long long)(E + Nn) * 32;
  long long lnT  = (long long)Nn * 32;
  long long nf   = (long long)Nn * 128;

  // --- layer 1
  gemm_wmma<128, 128, 4><<<gemmBlocks, blk, 0, stream>>>(x, W1, B0, Nn);
  fill_zero<<<cdiv(nf, BT), BT, 0, stream>>>(B1, nf);
  aggregate<128><<<cdiv(aggT, BT), BT, 0, stream>>>(B0, B1, srcI, dstI, dinv, E, Nn);
  ln_relu<<<cdiv(lnT, BT), BT, 0, stream>>>(B1, b1, g1v, be1, B0, Nn);

  // --- layer 2
  gemm_wmma<128, 128, 4><<<gemmBlocks, blk, 0, stream>>>(B0, W2, B1, Nn);
  fill_zero<<<cdiv(nf, BT), BT, 0, stream>>>(B0, nf);
  aggregate<128><<<cdiv(aggT, BT), BT, 0, stream>>>(B1, B0, srcI, dstI, dinv, E, Nn);
  ln_relu<<<cdiv(lnT, BT), BT, 0, stream>>>(B0, b2, g2v, be2, B1, Nn);

  // --- layer 3 (output width 64, aggregate straight into d_out)
  gemm_wmma<128, 64, 4><<<gemmBlocks, blk, 0, stream>>>(B1, W3, B0, Nn);
  long long of = (long long)Nn * 64;
  fill_zero<<<cdiv(of, BT), BT, 0, stream>>>(out, of);
  aggregate<64><<<cdiv(aggT, BT), BT, 0, stream>>>(B0, out, srcI, dstI, dinv, E, Nn);
  add_bias64<<<cdiv(of, BT), BT, 0, stream>>>(out, b3, of);
}